// MessagePassing_32796370273059
// MI455X (gfx1250) — compile-verified
//
#include <hip/hip_runtime.h>

// Scatter-add message passing: out[col[e]] += x[row[e]], E=500k, D=128, fp32.
// L2-resident; bound by L2 gather + atomic throughput. Mapping: one wave32 per
// 16-edge batch; within an edge, lane l covers floats [4l, 4l+4).

#define D_FEAT 128
#define BATCH  16   // edges per wave batch (one per lane 0..15)

// async-to-LDS builtin types: (v4i AS1* src_global, v4i AS3* dst_lds, imm, imm)
typedef int v4i_t __attribute__((vector_size(16)));
typedef __attribute__((address_space(1))) v4i_t* as1_v4i;
typedef __attribute__((address_space(3))) v4i_t* as3_v4i;

// Fire-and-forget device-scope fp32 atomics, saddr form: base pointer lives in
// an SGPR pair, per-edge address is one 32-bit voffset (out < 4 GiB). No
// return -> STOREcnt only; s_endpgm's implicit wait drains them. DEV scope:
// RMW must land at L2 (WGP$ is read-only). No "memory" clobber: out doesn't
// alias the __restrict__ inputs; lets loads hoist past the atomics.
__device__ __forceinline__ void atomic_add4(float* out_base, unsigned voff,
                                            float a, float b, float c, float d) {
  asm volatile(
      "global_atomic_add_f32 %0, %1, %5 offset:0  scope:SCOPE_DEV\n\t"
      "global_atomic_add_f32 %0, %2, %5 offset:4  scope:SCOPE_DEV\n\t"
      "global_atomic_add_f32 %0, %3, %5 offset:8  scope:SCOPE_DEV\n\t"
      "global_atomic_add_f32 %0, %4, %5 offset:12 scope:SCOPE_DEV"
      :
      : "v"(voff), "v"(a), "v"(b), "v"(c), "v"(d), "s"(out_base));
}

// Consume staged edge K: s_wait_asynccnt needs a frontend-constant, so this is
// template-recursive (async loads complete in order -> edge K is ready once
// <= BATCH-1-K remain outstanding; atomics pipeline against remaining gathers).
template <int K>
__device__ __forceinline__ void consume_from(const float* wave_lds, int my_c,
                                             float* out, int lane) {
  if constexpr (K < BATCH) {
    __builtin_amdgcn_s_wait_asynccnt(BATCH - 1 - K);
    const float4 v = *(const float4*)(wave_lds + K * D_FEAT + lane * 4);
    const int ci = __builtin_amdgcn_readlane(my_c, K);
    const unsigned voff =
        ((unsigned)ci * (D_FEAT * 4)) + ((unsigned)lane << 4);
    atomic_add4(out, voff, v.x, v.y, v.z, v.w);
    consume_from<K + 1>(wave_lds, my_c, out, lane);
  }
}

__global__ void __launch_bounds__(256)
scatter_add_edges(const float* __restrict__ x,
                  const long long* __restrict__ rows,
                  const long long* __restrict__ cols,
                  float* __restrict__ out,
                  int n_edges) {
  // Wave-private staging: 8 waves * 16 edges * 512B = 64 KB of the 320 KB LDS
  // -> 5 blocks (40 waves) per WGP.
  __shared__ __align__(16) float lds_buf[8 * BATCH * D_FEAT];
  float* wave_lds = lds_buf + (threadIdx.x >> 5) * (BATCH * D_FEAT);

  const int lane = threadIdx.x & 31;
  const int wv   = (int)((blockIdx.x * blockDim.x + threadIdx.x) >> 5);
  const int nw   = (int)((gridDim.x * blockDim.x) >> 5);

  const int n_batches = n_edges / BATCH;

  for (int b = wv; b < n_batches; b += nw) {
    const int base = b * BATCH;

    // Coalesced index fetch: lane k (k<16) owns edge base+k. readlane
    // broadcasts each edge's indices to SGPRs -> 2 vector loads per 16 edges
    // instead of 2 broadcast loads per edge.
    int my_r = 0, my_c = 0;
    if (lane < BATCH) {
      my_r = (int)rows[base + lane];
      my_c = (int)cols[base + lane];
    }
    // Prefetch the next stride of the index stream (global_prefetch_b8;
    // speculative OOB prefetches are dropped per ISA 10.5).
    __builtin_prefetch(&rows[base + nw * BATCH], 0, 1);
    __builtin_prefetch(&cols[base + nw * BATCH], 0, 1);

    // Stage all 16 scattered rows into LDS with async gathers: 16-deep MLP,
    // no VGPR pressure, tracked by ASYNCcnt instead of LOADcnt.
#pragma unroll
    for (int k = 0; k < BATCH; ++k) {
      const int ri = __builtin_amdgcn_readlane(my_r, k);
      const float* src = x + (size_t)ri * D_FEAT + (size_t)(lane * 4);
      float* dst = wave_lds + k * D_FEAT + lane * 4;
      __builtin_amdgcn_global_load_async_to_lds_b128(
          (as1_v4i)(__attribute__((address_space(1))) void*)src,
          (as3_v4i)(__attribute__((address_space(3))) void*)dst,
          /*offset=*/0, /*cpol=*/0);
    }
    // Drain in completion order, overlapping atomics with the gather tail.
    consume_from<0>(wave_lds, my_c, out, lane);
  }

  // Tail (< BATCH edges): wave 0 only, simple per-edge path.
  if (wv == 0) {
    for (int e = n_batches * BATCH; e < n_edges; ++e) {
      const int ri = __builtin_amdgcn_readfirstlane((int)rows[e]);
      const int ci = __builtin_amdgcn_readfirstlane((int)cols[e]);
      const float4 v = ((const float4*)(x + (size_t)ri * D_FEAT))[lane];
      const unsigned voff =
          ((unsigned)ci * (D_FEAT * 4)) + ((unsigned)lane << 4);
      atomic_add4(out, voff, v.x, v.y, v.z, v.w);
    }
  }
}

__global__ void __launch_bounds__(256)
zero_f4(float4* __restrict__ out, int n4) {
  int i = blockIdx.x * blockDim.x + threadIdx.x;
  if (i < n4) out[i] = make_float4(0.f, 0.f, 0.f, 0.f);
}

extern "C" void kernel_launch(void* const* d_in, const int* in_sizes, int n_in,
                              void* d_out, int out_size, void* d_ws, size_t ws_size,
                              hipStream_t stream) {
  (void)n_in; (void)d_ws; (void)ws_size;

  const float*     x  = (const float*)d_in[0];
  const long long* ei = (const long long*)d_in[1];  // int64 edge_index [2, E]
  const int n_edges   = in_sizes[1] / 2;
  const long long* rows = ei;             // edge_index[0] : source (gather)
  const long long* cols = ei + n_edges;   // edge_index[1] : target (scatter)
  float* out = (float*)d_out;

  // Harness poisons d_out with 0xAA -> zero it first (vectorized).
  const int n4 = out_size / 4;
  zero_f4<<<(n4 + 255) / 256, 256, 0, stream>>>((float4*)d_out, n4);

  // 2048 blocks * 8 waves = 16384 waves; ~2 batches of 16 edges per wave.
  scatter_add_edges<<<2048, 256, 0, stream>>>(x, rows, cols, out, n_edges);
}